// GCNNet_84817014161590
// MI455X (gfx1250) — compile-verified
//
#include <hip/hip_runtime.h>
#include <hip/hip_bf16.h>

typedef __attribute__((ext_vector_type(2))) float v2f;
typedef __attribute__((ext_vector_type(8))) float v8f;

#define NUM_GRAPHS 512
#define CH 128
#define APAD 4                      // LDS row padding (dwords) to dodge bank conflicts

// LDS byte-offset of a pointer into __shared__ memory (AS3 ptrtoint)
__device__ inline unsigned lds_off(const void* p) {
    return (unsigned)(size_t)(__attribute__((address_space(3))) const void*)p;
}

// ---------------------------------------------------------------------------
// fp32 WMMA GEMM: C[M,N] = A[M,K] @ B[K,N] (+bias) (+relu)
// One block = one 16-row M-strip; one wave per 16-col N-tile (blockDim = 2*N).
// A strip (16 x K) staged in LDS via GLOBAL_LOAD_ASYNC_TO_LDS_B128 (ASYNCcnt),
// then V_WMMA_F32_16X16X4_F32 with K stepped by 4, A from ds_load, B from L2.
// M,N multiples of 16; K multiple of 4; K <= 128.
// ---------------------------------------------------------------------------
__global__ __launch_bounds__(256) void wmma_gemm_f32(
    const float* __restrict__ A, const float* __restrict__ B,
    float* __restrict__ C, const float* __restrict__ bias,
    int M, int N, int K, int relu)
{
    __shared__ float As[16 * (CH + APAD)];

    const int tid  = threadIdx.x;
    const int wave = tid >> 5;          // N-tile index within block
    const int lane = tid & 31;
    const int half = lane >> 4;         // 0: lanes 0-15, 1: lanes 16-31
    const int l16  = lane & 15;
    const int m0   = blockIdx.x * 16;
    const int n0   = wave * 16;
    const int ldk  = K + APAD;          // padded LDS row stride (dwords)
    const int kq   = K >> 2;            // float4 chunks per row

    // ---- async stage of A strip: 16*K floats, b128 per chunk ----
    const int nchunks = 16 * kq;
    for (int c = tid; c < nchunks; c += blockDim.x) {
        const int row = c / kq;
        const int col = (c - row * kq) * 4;
        const float* gp = A + (size_t)(m0 + row) * K + col;
        unsigned loff = lds_off(&As[row * ldk + col]);
        asm volatile("global_load_async_to_lds_b128 %0, %1, off"
                     :: "v"(loff), "v"(gp) : "memory");
    }
    asm volatile("s_wait_asynccnt 0x0" ::: "memory");
    __syncthreads();

    // ---- WMMA main loop ----
    v8f acc = {};
    const float* bp0 = B + n0 + l16;
    for (int k0 = 0; k0 < K; k0 += 4) {
        const int kk = k0 + half * 2;   // A 16x4 layout: lanes 0-15 K=k0,k0+1;
        v2f a;                          //                lanes 16-31 K=k0+2,k0+3
        a.x = As[l16 * ldk + kk];
        a.y = As[l16 * ldk + kk + 1];
        v2f b;                          // B 4x16: rows striped across lanes
        const float* bp = bp0 + (size_t)kk * N;
        b.x = bp[0];
        b.y = bp[N];
        acc = __builtin_amdgcn_wmma_f32_16x16x4_f32(
            /*neg_a=*/false, a, /*neg_b=*/false, b,
            /*c_mod=*/(short)0, acc, /*reuse_a=*/false, /*reuse_b=*/false);
    }

    // ---- epilogue: C/D VGPR r -> M = m0 + r + half*8, N = n0 + l16 ----
    #pragma unroll
    for (int r = 0; r < 8; ++r) {
        const int m = m0 + r + half * 8;
        const int n = n0 + l16;
        float v = acc[r];
        if (bias) v += bias[n];
        if (relu) v = fmaxf(v, 0.0f);
        C[(size_t)m * N + n] = v;
    }
}

// ---------------------------------------------------------------------------
// Degree accumulation over edge targets (self-loop added later as +1)
// ---------------------------------------------------------------------------
__global__ void deg_kernel(const int* __restrict__ col, float* __restrict__ deg, int E)
{
    int e = blockIdx.x * blockDim.x + threadIdx.x;
    if (e < E) atomicAdd(&deg[col[e]], 1.0f);
}

__global__ void dinv_kernel(float* __restrict__ d, int N)
{
    int i = blockIdx.x * blockDim.x + threadIdx.x;
    if (i < N) d[i] = rsqrtf(d[i] + 1.0f);   // +1 self loop; always > 0
}

// ---------------------------------------------------------------------------
// Edge scatter: out[col[e], :] += dinv[row]*dinv[col] * h[row, :]
// 32 lanes per edge, float4 per lane (128 channels), coalesced 512B gather.
// ---------------------------------------------------------------------------
__global__ void edge_agg_kernel(const float* __restrict__ h, float* __restrict__ out,
                                const int* __restrict__ row, const int* __restrict__ col,
                                const float* __restrict__ dinv, int E)
{
    int t = blockIdx.x * blockDim.x + threadIdx.x;
    int e = t >> 5;
    if (e >= E) return;
    int lane = t & 31;
    int r = row[e], c = col[e];
    float nrm = dinv[r] * dinv[c];
    float4 v = ((const float4*)(h + (size_t)r * CH))[lane];
    float* op = out + (size_t)c * CH + lane * 4;
    atomicAdd(op + 0, nrm * v.x);
    atomicAdd(op + 1, nrm * v.y);
    atomicAdd(op + 2, nrm * v.z);
    atomicAdd(op + 3, nrm * v.w);
}

// ---------------------------------------------------------------------------
// Self-loop contribution + bias + ReLU (runs after edge_agg; unique writers)
// out[i,:] = relu(out[i,:] + dinv[i]^2 * h[i,:] + b)
// ---------------------------------------------------------------------------
__global__ void self_bias_relu_kernel(const float* __restrict__ h, float* __restrict__ out,
                                      const float* __restrict__ bias,
                                      const float* __restrict__ dinv, int N)
{
    int t = blockIdx.x * blockDim.x + threadIdx.x;
    int i = t >> 5;
    if (i >= N) return;
    int lane = t & 31;
    float d2 = dinv[i] * dinv[i];
    float4 hv = ((const float4*)(h + (size_t)i * CH))[lane];
    float4* op = (float4*)(out + (size_t)i * CH) + lane;
    float4 ov = *op;
    const float* b = bias + lane * 4;
    ov.x = fmaxf(ov.x + d2 * hv.x + b[0], 0.0f);
    ov.y = fmaxf(ov.y + d2 * hv.y + b[1], 0.0f);
    ov.z = fmaxf(ov.z + d2 * hv.z + b[2], 0.0f);
    ov.w = fmaxf(ov.w + d2 * hv.w + b[3], 0.0f);
    *op = ov;
}

// ---------------------------------------------------------------------------
// Global mean pool
// ---------------------------------------------------------------------------
__global__ void pool_sum_kernel(const float* __restrict__ h, const int* __restrict__ batch,
                                float* __restrict__ sums, int N)
{
    int t = blockIdx.x * blockDim.x + threadIdx.x;
    int i = t >> 5;
    if (i >= N) return;
    int lane = t & 31;
    int g = batch[i];
    float4 v = ((const float4*)(h + (size_t)i * CH))[lane];
    float* sp = sums + (size_t)g * CH + lane * 4;
    atomicAdd(sp + 0, v.x);
    atomicAdd(sp + 1, v.y);
    atomicAdd(sp + 2, v.z);
    atomicAdd(sp + 3, v.w);
}

__global__ void count_kernel(const int* __restrict__ batch, float* __restrict__ cnts, int N)
{
    int i = blockIdx.x * blockDim.x + threadIdx.x;
    if (i < N) atomicAdd(&cnts[batch[i]], 1.0f);
}

__global__ void pool_div_kernel(float* __restrict__ sums, const float* __restrict__ cnts)
{
    int t = blockIdx.x * blockDim.x + threadIdx.x;   // over 512*128
    if (t >= NUM_GRAPHS * CH) return;
    int g = t / CH;
    sums[t] = sums[t] / fmaxf(cnts[g], 1.0f);
}

// ---------------------------------------------------------------------------
extern "C" void kernel_launch(void* const* d_in, const int* in_sizes, int n_in,
                              void* d_out, int out_size, void* d_ws, size_t ws_size,
                              hipStream_t stream)
{
    // Inputs (setup_inputs order):
    const float* x    = (const float*)d_in[0];   // [N,128]
    const int*   ei   = (const int*)d_in[1];     // [2,E] flat
    const int*   batch= (const int*)d_in[2];     // [N]
    const float* W1   = (const float*)d_in[3];
    const float* b1   = (const float*)d_in[4];
    const float* W2   = (const float*)d_in[5];
    const float* b2   = (const float*)d_in[6];
    const float* W3   = (const float*)d_in[7];
    const float* b3   = (const float*)d_in[8];
    const float* Wm1  = (const float*)d_in[9];   // [128,64]
    const float* bm1  = (const float*)d_in[10];  // [64]
    const float* Wm2  = (const float*)d_in[11];  // [64,32]
    const float* bm2  = (const float*)d_in[12];  // [32]
    float* out = (float*)d_out;                  // [512,32]

    const int N = in_sizes[0] / CH;              // 100000
    const int E = in_sizes[1] / 2;               // 1600000
    const int* row = ei;         // sources
    const int* col = ei + E;     // targets

    // Workspace layout (floats)
    const size_t NH = (size_t)N * CH;
    float* H    = (float*)d_ws;          // GEMM output   [N,128]
    float* G    = H + NH;                // aggregated    [N,128]
    float* dinv = G + NH;                // [N]
    float* sums = dinv + N;              // [512,128]
    float* cnts = sums + (size_t)NUM_GRAPHS * CH;  // [512]
    float* z1   = cnts + NUM_GRAPHS;     // [512,64]

    const int TB = 256;
    const int ewarps_blocks = (E * 32 + TB - 1) / TB;   // edge kernel grid
    const int nwarps_blocks = (N * 32 + TB - 1) / TB;   // per-node float4 grid

    // ---- normalization ----
    hipMemsetAsync(dinv, 0, (size_t)N * sizeof(float), stream);
    deg_kernel<<<(E + TB - 1) / TB, TB, 0, stream>>>(col, dinv, E);
    dinv_kernel<<<(N + TB - 1) / TB, TB, 0, stream>>>(dinv, N);

    const float* layer_in = x;
    const float* Ws[3] = {W1, W2, W3};
    const float* bs[3] = {b1, b2, b3};

    for (int l = 0; l < 3; ++l) {
        // h = layer_in @ W   (block = 16-row strip, 8 waves = 8 N-tiles)
        wmma_gemm_f32<<<N / 16, 32 * (CH / 16), 0, stream>>>(layer_in, Ws[l], H,
                                                             nullptr, N, CH, CH, 0);
        // aggregate into G (zeroed after GEMM has consumed G's old contents)
        hipMemsetAsync(G, 0, NH * sizeof(float), stream);
        edge_agg_kernel<<<ewarps_blocks, TB, 0, stream>>>(H, G, row, col, dinv, E);
        self_bias_relu_kernel<<<nwarps_blocks, TB, 0, stream>>>(H, G, bs[l], dinv, N);
        layer_in = G;
    }

    // ---- global mean pool ----
    hipMemsetAsync(sums, 0, (size_t)NUM_GRAPHS * CH * sizeof(float), stream);
    hipMemsetAsync(cnts, 0, (size_t)NUM_GRAPHS * sizeof(float), stream);
    pool_sum_kernel<<<nwarps_blocks, TB, 0, stream>>>(G, batch, sums, N);
    count_kernel<<<(N + TB - 1) / TB, TB, 0, stream>>>(batch, cnts, N);
    pool_div_kernel<<<(NUM_GRAPHS * CH + TB - 1) / TB, TB, 0, stream>>>(sums, cnts);

    // ---- MLP head ----
    wmma_gemm_f32<<<NUM_GRAPHS / 16, 32 * (64 / 16), 0, stream>>>(
        sums, Wm1, z1, bm1, NUM_GRAPHS, 64, CH, 1);
    wmma_gemm_f32<<<NUM_GRAPHS / 16, 32 * (32 / 16), 0, stream>>>(
        z1, Wm2, out, bm2, NUM_GRAPHS, 32, 64, 0);
    (void)n_in; (void)out_size; (void)ws_size;
}